// InterleavedUpsampleExample_28020366639577
// MI455X (gfx1250) — compile-verified
//
#include <hip/hip_runtime.h>

// InterleavedUpsample: out[b, 2i+kh, 2j+kw] = {x0, x1, x2, -1}[kh*2+kw]
// Pure streaming permutation: 192 MiB in + 256 MiB out, 0 FLOPs.
// Optimal MI455X strategy: fully-coalesced b128 loads/stores with NT temporal
// hints (touch-once data, don't rinse the 192MB L2) + gfx1250 global prefetch.

typedef __attribute__((ext_vector_type(4))) float v4f;

#define BDIM 16
#define CDIM 3
#define HDIM 1024
#define WDIM 1024
#define W4   (WDIM / 4)                       // 256 float4-tiles per input row
#define NTILES ((long)BDIM * HDIM * W4)       // 4,194,304 tiles total

__global__ __launch_bounds__(256) void interleave_upsample_kernel(
    const float* __restrict__ x, float* __restrict__ out)
{
    const long stride = (long)gridDim.x * blockDim.x;
    long idx = (long)blockIdx.x * blockDim.x + threadIdx.x;

    for (; idx < NTILES; idx += stride) {
        // Decode (b, i, j4). All divisors are powers of two -> shifts/masks.
        const int  j4  = (int)(idx & (W4 - 1));          // tile within row
        const long row = idx >> 8;                       // b*H + i
        const int  i   = (int)(row & (HDIM - 1));
        const int  b   = (int)(row >> 10);

        const long chan_stride = (long)HDIM * WDIM;
        const long in_base = (((long)b * CDIM) * HDIM + i) * WDIM + 4L * j4;

        // Prefetch next iteration's three input vectors (exact addresses).
        const long nidx = idx + stride;
        if (nidx < NTILES) {
            const int  nj4  = (int)(nidx & (W4 - 1));
            const long nrow = nidx >> 8;
            const int  ni   = (int)(nrow & (HDIM - 1));
            const int  nb   = (int)(nrow >> 10);
            const long nbase = (((long)nb * CDIM) * HDIM + ni) * WDIM + 4L * nj4;
            __builtin_prefetch(x + nbase,                   0, 0);
            __builtin_prefetch(x + nbase + chan_stride,     0, 0);
            __builtin_prefetch(x + nbase + 2 * chan_stride, 0, 0);
        }

        // Touch-once streaming loads (TH=NT).
        const v4f x0 = __builtin_nontemporal_load((const v4f*)(x + in_base));
        const v4f x1 = __builtin_nontemporal_load((const v4f*)(x + in_base + chan_stride));
        const v4f x2 = __builtin_nontemporal_load((const v4f*)(x + in_base + 2 * chan_stride));

        // Interleave in VGPRs (free next to 448 MiB of HBM traffic).
        v4f e0 = { x0.x, x1.x, x0.y, x1.y };             // even row, cols 8j..8j+3
        v4f e1 = { x0.z, x1.z, x0.w, x1.w };             // even row, cols 8j+4..8j+7
        v4f o0 = { x2.x, -1.0f, x2.y, -1.0f };           // odd  row, cols 8j..8j+3
        v4f o1 = { x2.z, -1.0f, x2.w, -1.0f };           // odd  row, cols 8j+4..8j+7

        // out shape (B, 1, 2H, 2W); rows 2i and 2i+1, 8 contiguous floats each.
        const long out_base = (((long)b * (2 * HDIM) + 2 * i) * (2L * WDIM)) + 8L * j4;
        float* oe = out + out_base;
        float* oo = oe + 2 * WDIM;
        __builtin_nontemporal_store(e0, (v4f*)(oe));
        __builtin_nontemporal_store(e1, (v4f*)(oe + 4));
        __builtin_nontemporal_store(o0, (v4f*)(oo));
        __builtin_nontemporal_store(o1, (v4f*)(oo + 4));
    }
}

extern "C" void kernel_launch(void* const* d_in, const int* in_sizes, int n_in,
                              void* d_out, int out_size, void* d_ws, size_t ws_size,
                              hipStream_t stream) {
    (void)in_sizes; (void)n_in; (void)out_size; (void)d_ws; (void)ws_size;
    const float* x  = (const float*)d_in[0];
    float* out      = (float*)d_out;

    // 8192 blocks x 256 threads (8 waves/block, wave32): 2,097,152 threads,
    // exactly 2 tiles per thread -> the grid-stride loop runs twice and the
    // prefetch overlaps iteration 2's loads with iteration 1's stores.
    dim3 grid(8192), block(256);
    hipLaunchKernelGGL(interleave_upsample_kernel, grid, block, 0, stream, x, out);
}